// VectorQuantizer_43215960933207
// MI455X (gfx1250) — compile-verified
//
#include <hip/hip_runtime.h>

typedef __attribute__((ext_vector_type(2))) float v2f;
typedef __attribute__((ext_vector_type(8))) float v8f;

#define NUM_EMB   512
#define DIM       64
#define HW        4096          // 64*64
#define CHW       262144        // DIM*HW
#define NUMEL     8388608u      // 32*64*64*64
#define NBLOCKS   512
#define WPB       8             // waves per block (256 threads)
#define NWAVES    (NBLOCKS*WPB) // 4096 waves, 2 tiles each -> 8192 tiles * 16 = 131072 pts

__global__ __launch_bounds__(256) void vq_main(const float* __restrict__ in,
                                               const float* __restrict__ cb,
                                               float* __restrict__ out,
                                               double* __restrict__ partial)
{
    __shared__ float norms[NUM_EMB];
    __shared__ int   idx_lds[WPB * 32];

    const int tid = threadIdx.x;

    // --- codebook squared norms (cooperative, cached reads) ---
    for (int k = tid; k < NUM_EMB; k += 256) {
        const float4* row = (const float4*)(cb + k * DIM);
        float s = 0.0f;
        #pragma unroll
        for (int c4 = 0; c4 < DIM / 4; ++c4) {
            float4 v = row[c4];
            s += v.x * v.x + v.y * v.y + v.z * v.z + v.w * v.w;
        }
        norms[k] = s;
    }
    __syncthreads();

    const int lane = tid & 31;
    const int wave = tid >> 5;
    const int wgid = blockIdx.x * WPB + wave;
    const int m    = lane & 15;       // row within 16-pt / 16-code tile
    const int hi   = lane >> 4;       // half-wave select
    const int fl   = 2 * hi;          // K-offset of this half per WMMA step

    // two point tiles per wave
    const int n0p = (wgid * 2 + 0) * 16 + m;
    const int n1p = (wgid * 2 + 1) * 16 + m;
    const long pbase0 = (long)(n0p >> 12) * CHW + (n0p & 4095);
    const long pbase1 = (long)(n1p >> 12) * CHW + (n1p & 4095);

    // --- load A (16 pts x 64 feats, twice): lane l vgpr j <- A[m][4s+fl+j] ---
    v2f a0[16], a1[16];
    #pragma unroll
    for (int s = 0; s < 16; ++s) {
        const int f = 4 * s + fl;
        a0[s].x = in[pbase0 + (long)f * HW];
        a0[s].y = in[pbase0 + (long)(f + 1) * HW];
        a1[s].x = in[pbase1 + (long)f * HW];
        a1[s].y = in[pbase1 + (long)(f + 1) * HW];
    }

    float bv0[8], bv1[8];
    int   bi0[8], bi1[8];
    #pragma unroll
    for (int v = 0; v < 8; ++v) { bv0[v] = 3.4e38f; bv1[v] = 3.4e38f; bi0[v] = 0; bi1[v] = 0; }

    const float* cb_lane = cb + m * DIM + fl;   // B[k_row][code=m], k_row = fl(+j)

    // --- sweep all 512 codes in 16-wide tiles; 16 f32 WMMA per tile pair ---
    for (int nt = 0; nt < NUM_EMB / 16; ++nt) {
        const int n0 = nt * 16;
        const float sn = norms[n0 + m];          // broadcast per code column
        v8f acc0 = {}; v8f acc1 = {};
        const v2f* bp = (const v2f*)(cb_lane + n0 * DIM);
        #pragma unroll
        for (int s = 0; s < 16; ++s) {
            v2f b = bp[2 * s];                   // cb[n0+m][4s+fl .. +1]
            acc0 = __builtin_amdgcn_wmma_f32_16x16x4_f32(false, a0[s], false, b,
                                                         (short)0, acc0, false, false);
            acc1 = __builtin_amdgcn_wmma_f32_16x16x4_f32(false, a1[s], false, b,
                                                         (short)0, acc1, false, false);
        }
        const int code = n0 + m;
        #pragma unroll
        for (int v = 0; v < 8; ++v) {
            float s0 = sn - 2.0f * acc0[v];
            if (s0 < bv0[v]) { bv0[v] = s0; bi0[v] = code; }
            float s1 = sn - 2.0f * acc1[v];
            if (s1 < bv1[v]) { bv1[v] = s1; bi1[v] = code; }
        }
    }

    // --- argmin across the 16 code-columns of each half-wave ---
    #pragma unroll
    for (int v = 0; v < 8; ++v) {
        #pragma unroll
        for (int off = 8; off >= 1; off >>= 1) {
            float ov = __shfl_xor(bv0[v], off, 16);
            int   oi = __shfl_xor(bi0[v], off, 16);
            if (ov < bv0[v] || (ov == bv0[v] && oi < bi0[v])) { bv0[v] = ov; bi0[v] = oi; }
            ov = __shfl_xor(bv1[v], off, 16);
            oi = __shfl_xor(bi1[v], off, 16);
            if (ov < bv1[v] || (ov == bv1[v] && oi < bi1[v])) { bv1[v] = ov; bi1[v] = oi; }
        }
    }

    // lanes 0-15 of vgpr v hold point M=v; lanes 16-31 hold point M=v+8
    int* widx = idx_lds + wave * 32;
    if (m == 0) {
        #pragma unroll
        for (int v = 0; v < 8; ++v) {
            widx[hi * 8 + v]      = bi0[v];
            widx[16 + hi * 8 + v] = bi1[v];
        }
    }
    __syncthreads();

    const int q0 = widx[m];
    const int q1 = widx[16 + m];
    const float* c0 = cb + q0 * DIM;
    const float* c1 = cb + q1 * DIM;

    double lsum = 0.0;
    #pragma unroll
    for (int s = 0; s < 16; ++s) {
        const int f = 4 * s + fl;
        float qa = c0[f], qb = c0[f + 1];
        float d0 = qa - a0[s].x;
        float d1 = qb - a0[s].y;
        lsum += (double)(d0 * d0) + (double)(d1 * d1);
        out[pbase0 + (long)f * HW]       = a0[s].x + d0;   // straight-through value
        out[pbase0 + (long)(f + 1) * HW] = a0[s].y + d1;
        qa = c1[f]; qb = c1[f + 1];
        d0 = qa - a1[s].x;
        d1 = qb - a1[s].y;
        lsum += (double)(d0 * d0) + (double)(d1 * d1);
        out[pbase1 + (long)f * HW]       = a1[s].x + d0;
        out[pbase1 + (long)(f + 1) * HW] = a1[s].y + d1;
    }

    // --- wave-level loss reduction; deterministic partial per wave ---
    #pragma unroll
    for (int off = 16; off >= 1; off >>= 1)
        lsum += __shfl_xor(lsum, off, 32);
    if (lane == 0) partial[wgid] = lsum;
}

__global__ __launch_bounds__(256) void vq_loss_reduce(const double* __restrict__ partial,
                                                      float* __restrict__ out_loss)
{
    __shared__ double sh[256];
    double s = 0.0;
    for (int i = threadIdx.x; i < NWAVES; i += 256) s += partial[i];
    sh[threadIdx.x] = s;
    __syncthreads();
    if (threadIdx.x == 0) {
        double t = 0.0;
        for (int i = 0; i < 256; ++i) t += sh[i];   // fixed order -> deterministic
        *out_loss = (float)(1.25 * t / (double)NUMEL);
    }
}

extern "C" void kernel_launch(void* const* d_in, const int* in_sizes, int n_in,
                              void* d_out, int out_size, void* d_ws, size_t ws_size,
                              hipStream_t stream) {
    const float* in = (const float*)d_in[0];   // (32, 64, 64, 64) f32
    const float* cb = (const float*)d_in[1];   // (512, 64) f32
    float* out      = (float*)d_out;           // 8388608 quantized_st + 1 loss
    double* partial = (double*)d_ws;           // NWAVES doubles (32 KB)

    vq_main<<<NBLOCKS, 256, 0, stream>>>(in, cb, out, partial);
    vq_loss_reduce<<<1, 256, 0, stream>>>(partial, out + NUMEL);
}